// PSOLAPitchShifter_23948737642968
// MI455X (gfx1250) — compile-verified
//
#include <hip/hip_runtime.h>
#include <hip/hip_bf16.h>
#include <cstdint>

// PSOLA pitch shifter, gather formulation (no atomics).
// FRAME_LEN = 1024, HOP = 256. Each output sample t is covered by <=4 frames
// f in [ceil((t-1023)/256), t/256]. Per block of 256 samples, the accessed
// wav window is 1792 floats, staged into LDS via CDNA5 async-to-LDS loads.

#define PS_L    1024
#define PS_HOP  256
#define PS_WIN  1792            // (3 + 4) * 256 floats staged per block

#if __has_builtin(__builtin_amdgcn_global_load_async_to_lds_b128)
#define PS_HAS_ASYNC_LDS 1
#else
#define PS_HAS_ASYNC_LDS 0
#endif

// Builtin signature (from hipcc diagnostic): both pointer args are
// 'int __vector(4) *' with address-space qualifiers (1 = global, 3 = LDS).
typedef int ps_int4 __attribute__((vector_size(16)));
typedef __attribute__((address_space(1))) ps_int4* ps_gp128;
typedef __attribute__((address_space(3))) ps_int4* ps_lp128;

// ---------------------------------------------------------------------------
// Kernel 1: per-frame parameters.
//   pk[f] = new_len if frame is valid (voiced && src>=1 && tgt>=1), else -1
//   sc[f] = 1024.0f / new_len
// ---------------------------------------------------------------------------
__global__ __launch_bounds__(256) void psola_params_kernel(
    const float* __restrict__ src_f0,
    const float* __restrict__ tgt_f0,
    const unsigned char* __restrict__ voiced,   // jax bool -> 1 byte/elem
    float* __restrict__ sc,
    int* __restrict__ pk,
    int nf)
{
    int f = blockIdx.x * blockDim.x + threadIdx.x;
    if (f >= nf) return;
    float s = src_f0[f];
    float t = tgt_f0[f];
    bool valid = (voiced[f] != 0) && (s >= 1.0f) && (t >= 1.0f);
    float denom = (t >= 1.0f) ? t : 1.0f;
    float ratio = fminf(fmaxf(s / denom, 0.25f), 4.0f);
    float nlf   = fmaxf(1.0f, rintf(1024.0f * ratio));   // round-half-even like jnp.round
    int   nl    = (int)nlf;
    sc[f] = 1024.0f / (float)nl;
    pk[f] = valid ? nl : -1;
}

// ---------------------------------------------------------------------------
// Kernel 2: main PSOLA gather / overlap-add / normalize.
// ---------------------------------------------------------------------------
__global__ __launch_bounds__(256) void psola_main_kernel(
    const float* __restrict__ wav,
    const float* __restrict__ sc,
    const int*   __restrict__ pk,
    float* __restrict__ out,
    int T, int nf)
{
    __shared__ alignas(16) float lwav[PS_WIN];

    const int b   = blockIdx.x;
    const int tid = threadIdx.x;
    const int t   = b * 256 + tid;
    const long long base_e = (long long)(b - 3) * 256;   // wav element index of lwav[0]

    // ---- Stage wav[(b-3)*256 .. b*256+1023] into LDS (clamped at array ends).
#if PS_HAS_ASYNC_LDS
    {
        // 1792 floats = 448 x b128 chunks; 256 threads -> chunks tid and tid+256.
        long long g0 = base_e + (long long)tid * 4;
        if (g0 < 0) g0 = 0;
        if (g0 > (long long)T - 4) g0 = (long long)T - 4;
        __builtin_amdgcn_global_load_async_to_lds_b128(
            (ps_gp128)(uintptr_t)(wav + g0),
            (ps_lp128)(uint32_t)(uintptr_t)(&lwav[tid * 4]),
            0, 0);
        if (tid < PS_WIN / 4 - 256) {   // tid < 192
            long long g1 = base_e + (long long)(tid + 256) * 4;
            if (g1 < 0) g1 = 0;
            if (g1 > (long long)T - 4) g1 = (long long)T - 4;
            __builtin_amdgcn_global_load_async_to_lds_b128(
                (ps_gp128)(uintptr_t)(wav + g1),
                (ps_lp128)(uint32_t)(uintptr_t)(&lwav[(tid + 256) * 4]),
                0, 0);
        }
#if __has_builtin(__builtin_amdgcn_s_wait_asynccnt)
        __builtin_amdgcn_s_wait_asynccnt(0);
#endif
    }
#else
    for (int k = tid; k < PS_WIN; k += 256) {
        long long g = base_e + k;
        if (g < 0) g = 0;
        if (g > (long long)T - 1) g = (long long)T - 1;
        lwav[k] = wav[g];
    }
#endif
    __syncthreads();

    if (t >= T) return;

    // wav[t] lives at lwav[tid + 768] (t - base_e = tid + 768).
    const float wav_t = lwav[tid + 768];

    int f_hi = t >> 8;                       // floor(t / 256)
    if (f_hi > nf - 1) f_hi = nf - 1;
    int f_lo = (t >= PS_L) ? (((t - PS_L) >> 8) + 1) : 0;

    float accO = 0.0f;
    float accW = 0.0f;

    for (int f = f_lo; f <= f_hi; ++f) {
        int j = t - (f << 8);                                // position in frame, 0..1023
        float h = 0.5f - 0.5f * __cosf((float)j * 6.1359233e-3f); // 2*pi/1024
        int nl = pk[f];
        float s;
        if (nl >= 0) {                                       // valid (voiced) frame
            if (j < nl) {
                float scale = sc[f];
                float x = ((float)j + 0.5f) * scale - 0.5f;
                x = fminf(fmaxf(x, 0.0f), 1023.0f);
                int   x0 = (int)x;                           // floor, x >= 0
                float w  = x - (float)x0;
                int   x1 = x0 + 1; if (x1 > 1023) x1 = 1023;
                int   fb = (f - b + 3) << 8;                 // frame start inside lwav
                float v0 = lwav[fb + x0];
                float v1 = lwav[fb + x1];
                s = v0 * (1.0f - w) + v1 * w;
            } else {
                s = 0.0f;                                    // masked, but window still counts
            }
        } else {
            s = wav_t;                                       // unvoiced: original frame sample
        }
        accO += s * h;
        accW += h;
    }

    out[t] = (accW > 1e-8f) ? (accO / accW) : accO;
}

// ---------------------------------------------------------------------------
// Launch: inputs in setup_inputs() order: wav, src_f0, tgt_f0, voiced_mask.
// ---------------------------------------------------------------------------
extern "C" void kernel_launch(void* const* d_in, const int* in_sizes, int n_in,
                              void* d_out, int out_size, void* d_ws, size_t ws_size,
                              hipStream_t stream)
{
    const float*         wav    = (const float*)d_in[0];
    const float*         src_f0 = (const float*)d_in[1];
    const float*         tgt_f0 = (const float*)d_in[2];
    const unsigned char* voiced = (const unsigned char*)d_in[3];
    float*               out    = (float*)d_out;

    const int T = in_sizes[0];
    int nf = in_sizes[1];
    const int nf_cap = (T - PS_L) / PS_HOP + 1;
    if (nf > nf_cap) nf = nf_cap;

    const int nfp = (nf + 255) & ~255;
    float* sc = (float*)d_ws;
    int*   pk = (int*)((char*)d_ws + (size_t)nfp * sizeof(float));

    psola_params_kernel<<<(nf + 255) / 256, 256, 0, stream>>>(
        src_f0, tgt_f0, voiced, sc, pk, nf);

    const int nblocks = (T + 255) / 256;
    psola_main_kernel<<<nblocks, 256, 0, stream>>>(wav, sc, pk, out, T, nf);
}